// GCNEdge2Cluster_11321533792258
// MI455X (gfx1250) — compile-verified
//
#include <hip/hip_runtime.h>
#include <hip/hip_bf16.h>

typedef __attribute__((ext_vector_type(2))) float v2f;
typedef __attribute__((ext_vector_type(8))) float v8f;

#define IN_DIM 128
#define HID 32
#define NCLUS 30
#define PADC 32

// ---------------- init: deg=1, padded W2, zero colsum/sse ----------------
__global__ void init_k(const float* __restrict__ W2, float* __restrict__ deg,
                       float* __restrict__ W2p, float* __restrict__ acc, int n) {
  int t = blockIdx.x * blockDim.x + threadIdx.x;
  if (t < n) deg[t] = 1.0f;                       // self-loop contributes 1
  if (t < HID * PADC) {
    int r = t >> 5, c = t & 31;
    W2p[t] = (c < NCLUS) ? W2[r * NCLUS + c] : 0.0f;
  }
  if (t < 33) acc[t] = 0.0f;                      // colsum[0..31], sse at [32]
}

// ---------------- degree accumulation over dst ----------------
__global__ void deg_k(const int* __restrict__ dst, float* __restrict__ deg, int nE) {
  int e = blockIdx.x * blockDim.x + threadIdx.x;
  if (e < nE) atomicAdd(&deg[dst[e]], 1.0f);
}

__global__ void dinv_k(float* __restrict__ deg, int n) {
  int t = blockIdx.x * blockDim.x + threadIdx.x;
  if (t < n) deg[t] = rsqrtf(deg[t]);             // deg >= 1 always (self loop)
}

// ---------------- GEMM1: h1 = x @ W1 ([N,128]@[128,32]) via fp32 WMMA ------------
// One wave per 16-row strip -> full 16x32 output: A fragment reused for both
// column tiles; epilogue fuses seed: out1 = b1 + dinv^2 * h1.
__global__ void gemm1_k(const float* __restrict__ x, const float* __restrict__ W1,
                        const float* __restrict__ dinv, const float* __restrict__ b1,
                        float* __restrict__ h1, float* __restrict__ out1, int nstrips) {
  int wave = (blockIdx.x * blockDim.x + threadIdx.x) >> 5;
  if (wave >= nstrips) return;                    // wave-uniform; EXEC stays all-1
  int lane = threadIdx.x & 31;
  int l15  = lane & 15;
  int m0   = wave * 16;
  int mrow = m0 + l15;                            // A layout: M = lane&15
  int kp   = (lane >> 4) * 2;                     // K-pair select per ISA A layout
  const float* __restrict__ xrow = x + (size_t)mrow * IN_DIM;
  v8f acc0 = {};                                  // cols 0..15
  v8f acc1 = {};                                  // cols 16..31
  #pragma unroll
  for (int k = 0; k < IN_DIM; k += 4) {
    v2f a, b0, b1f;
    a.x  = xrow[k + kp];
    a.y  = xrow[k + kp + 1];
    b0.x = W1[(k + kp) * HID + l15];
    b0.y = W1[(k + kp + 1) * HID + l15];
    b1f.x = W1[(k + kp) * HID + 16 + l15];
    b1f.y = W1[(k + kp + 1) * HID + 16 + l15];
    acc0 = __builtin_amdgcn_wmma_f32_16x16x4_f32(false, a, false, b0, (short)0, acc0,
                                                 false, false);
    acc1 = __builtin_amdgcn_wmma_f32_16x16x4_f32(false, a, false, b1f, (short)0, acc1,
                                                 false, false);
  }
  float bias0 = b1[l15];
  float bias1 = b1[16 + l15];
  int mbase = m0 + ((lane >> 4) << 3);            // C/D layout: M = r + 8*lane[4]
  #pragma unroll
  for (int r = 0; r < 8; ++r) {
    int row = mbase + r;
    float hv0 = acc0[r], hv1 = acc1[r];
    float di = dinv[row];
    float d2 = di * di;
    h1[(size_t)row * PADC + l15]        = hv0;
    h1[(size_t)row * PADC + 16 + l15]   = hv1;
    out1[(size_t)row * PADC + l15]      = bias0 + d2 * hv0;
    out1[(size_t)row * PADC + 16 + l15] = bias1 + d2 * hv1;
  }
}

// ---------------- edge scatter: out[dst] += dinv[s]*dinv[d]*h[src] (wave/edge) ----
__global__ void scatter_k(const int* __restrict__ src, const int* __restrict__ dst,
                          const float* __restrict__ dinv, const float* __restrict__ h,
                          float* __restrict__ out, int nE) {
  long long tid = (long long)blockIdx.x * blockDim.x + threadIdx.x;
  int e = (int)(tid >> 5);
  if (e >= nE) return;
  int c = (int)(tid & 31);
  int s = src[e], d = dst[e];                     // uniform within wave -> scalarized
  float norm = dinv[s] * dinv[d];
  atomicAdd(&out[(size_t)d * PADC + c], norm * h[(size_t)s * PADC + c]);
}

// ---------------- GEMM2: logits = relu(out1) @ W2p ([N,32]@[32,32]) --------------
// Fused ReLU on A operand; epilogue seeds out2 = b2p + dinv^2 * logits.
__global__ void gemm2_k(const float* __restrict__ hin, const float* __restrict__ W2p,
                        const float* __restrict__ dinv, const float* __restrict__ b2,
                        float* __restrict__ logits, float* __restrict__ out2,
                        int nstrips) {
  int wave = (blockIdx.x * blockDim.x + threadIdx.x) >> 5;
  if (wave >= nstrips) return;
  int lane = threadIdx.x & 31;
  int l15  = lane & 15;
  int m0   = wave * 16;
  int mrow = m0 + l15;
  int kp   = (lane >> 4) * 2;
  const float* __restrict__ row = hin + (size_t)mrow * HID;
  v8f acc0 = {};
  v8f acc1 = {};
  #pragma unroll
  for (int k = 0; k < HID; k += 4) {
    v2f a, b0, b1f;
    a.x  = fmaxf(row[k + kp], 0.0f);              // fused ReLU
    a.y  = fmaxf(row[k + kp + 1], 0.0f);
    b0.x = W2p[(k + kp) * PADC + l15];
    b0.y = W2p[(k + kp + 1) * PADC + l15];
    b1f.x = W2p[(k + kp) * PADC + 16 + l15];
    b1f.y = W2p[(k + kp + 1) * PADC + 16 + l15];
    acc0 = __builtin_amdgcn_wmma_f32_16x16x4_f32(false, a, false, b0, (short)0, acc0,
                                                 false, false);
    acc1 = __builtin_amdgcn_wmma_f32_16x16x4_f32(false, a, false, b1f, (short)0, acc1,
                                                 false, false);
  }
  float bias0 = (l15 < NCLUS) ? b2[l15] : 0.0f;
  float bias1 = (16 + l15 < NCLUS) ? b2[16 + l15] : 0.0f;
  int mbase = m0 + ((lane >> 4) << 3);
  #pragma unroll
  for (int r = 0; r < 8; ++r) {
    int rw = mbase + r;
    float lv0 = acc0[r], lv1 = acc1[r];
    float di = dinv[rw];
    float d2 = di * di;
    logits[(size_t)rw * PADC + l15]      = lv0;
    logits[(size_t)rw * PADC + 16 + l15] = lv1;
    out2[(size_t)rw * PADC + l15]        = bias0 + d2 * lv0;
    out2[(size_t)rw * PADC + 16 + l15]   = bias1 + d2 * lv1;
  }
}

// ---------------- softmax (c<30) -> FX, plus column sums of log(1-FX^2) ----------
__global__ void softmax_k(const float* __restrict__ out2, float* __restrict__ fx,
                          float* __restrict__ colsum, int n) {
  __shared__ float s_col[NCLUS];
  int t = threadIdx.x;
  if (t < NCLUS) s_col[t] = 0.0f;
  __syncthreads();
  int node = blockIdx.x * blockDim.x + t;
  if (node < n) {
    const float* __restrict__ row = out2 + (size_t)node * PADC;
    float m = row[0];
    #pragma unroll
    for (int c = 1; c < NCLUS; ++c) m = fmaxf(m, row[c]);
    float ex[NCLUS];
    float sum = 0.0f;
    #pragma unroll
    for (int c = 0; c < NCLUS; ++c) { ex[c] = expf(row[c] - m); sum += ex[c]; }
    float inv = 1.0f / sum;
    #pragma unroll
    for (int c = 0; c < NCLUS; ++c) {
      float p = ex[c] * inv;
      fx[(size_t)node * NCLUS + c] = p;
      atomicAdd(&s_col[c], logf(1.0f - p * p));   // LDS float atomic
    }
  }
  __syncthreads();
  if (t < NCLUS) atomicAdd(&colsum[t], s_col[t]);
}

// ---------------- FF = <FX[src],FX[dst]>; SSE accumulation ----------------
__global__ void ff_k(const int* __restrict__ src, const int* __restrict__ dst,
                     const float* __restrict__ fx, const float* __restrict__ epred,
                     float* __restrict__ sse, int nE) {
  int e = blockIdx.x * blockDim.x + threadIdx.x;
  float v = 0.0f;
  if (e < nE) {
    int s = src[e], d = dst[e];
    const float* __restrict__ a = fx + (size_t)s * NCLUS;
    const float* __restrict__ b = fx + (size_t)d * NCLUS;
    float dot = 0.0f;
    #pragma unroll
    for (int c = 0; c < NCLUS; ++c) dot += a[c] * b[c];
    float diff = dot - epred[e];
    v = diff * diff;
  }
  __shared__ float red[256];
  red[threadIdx.x] = v;
  __syncthreads();
  for (int w = 128; w > 0; w >>= 1) {
    if (threadIdx.x < w) red[threadIdx.x] += red[threadIdx.x + w];
    __syncthreads();
  }
  if (threadIdx.x == 0) atomicAdd(sse, red[0]);
}

// ---------------- finalize: preg + loss ----------------
__global__ void fin_k(const float* __restrict__ colsum, const float* __restrict__ sse,
                      float* __restrict__ loss_out, float invE) {
  float preg = 0.0f;
  #pragma unroll
  for (int c = 0; c < NCLUS; ++c)
    preg -= logf(1.0001f - expf(colsum[c]));
  *loss_out = (*sse) * invE + 0.01f * preg;
}

extern "C" void kernel_launch(void* const* d_in, const int* in_sizes, int n_in,
                              void* d_out, int out_size, void* d_ws, size_t ws_size,
                              hipStream_t stream) {
  const float* x     = (const float*)d_in[0];
  const int*   eidx  = (const int*)d_in[1];
  const float* epred = (const float*)d_in[2];
  const float* W1    = (const float*)d_in[3];
  const float* b1    = (const float*)d_in[4];
  const float* W2    = (const float*)d_in[5];
  const float* b2    = (const float*)d_in[6];

  const int N  = in_sizes[0] / IN_DIM;   // 100000 (multiple of 16)
  const int E  = in_sizes[1] / 2;        // 3200000
  const int* src = eidx;
  const int* dst = eidx + E;

  float* fx   = (float*)d_out;           // [N,30]
  float* loss = fx + (size_t)N * NCLUS;  // scalar at the end

  // workspace layout (floats), buffers aliased to stay small (~39 MB)
  float* ws    = (float*)d_ws;
  size_t off   = 0;
  float* dinv  = ws + off; off += ((size_t)N + 63) & ~(size_t)63;   // deg -> dinv in place
  float* bufA  = ws + off; off += (size_t)N * PADC;                 // h1, later out2
  float* bufB  = ws + off; off += (size_t)N * PADC;                 // out1 (pre-relu)
  float* bufC  = ws + off; off += (size_t)N * PADC;                 // logits
  float* W2p   = ws + off; off += HID * PADC;                       // padded W2 [32,32]
  float* acc   = ws + off; off += 64;                               // colsum[0..31], sse[32]
  float* colsum = acc;
  float* sse    = acc + 32;

  const int nstrips = N / 16;            // one wave per 16x32 output strip
  const long long edgeChan = (long long)E * PADC;

  init_k<<<(N + 255) / 256, 256, 0, stream>>>(W2, dinv, W2p, acc, N);
  deg_k<<<(E + 255) / 256, 256, 0, stream>>>(dst, dinv, E);
  dinv_k<<<(N + 255) / 256, 256, 0, stream>>>(dinv, N);

  gemm1_k<<<(nstrips * 32 + 127) / 128, 128, 0, stream>>>(x, W1, dinv, b1,
                                                          bufA, bufB, nstrips);
  scatter_k<<<(unsigned)((edgeChan + 255) / 256), 256, 0, stream>>>(src, dst, dinv,
                                                                    bufA, bufB, E);

  gemm2_k<<<(nstrips * 32 + 127) / 128, 128, 0, stream>>>(bufB, W2p, dinv, b2,
                                                          bufC, bufA, nstrips);
  scatter_k<<<(unsigned)((edgeChan + 255) / 256), 256, 0, stream>>>(src, dst, dinv,
                                                                    bufC, bufA, E);

  softmax_k<<<(N + 255) / 256, 256, 0, stream>>>(bufA, fx, colsum, N);
  ff_k<<<(E + 255) / 256, 256, 0, stream>>>(src, dst, fx, epred, sse, E);
  fin_k<<<1, 1, 0, stream>>>(colsum, sse, loss, 1.0f / (float)E);
}